// SSIMLoss_89361089561400
// MI455X (gfx1250) — compile-verified
//
#include <hip/hip_runtime.h>
#include <math.h>

typedef __attribute__((ext_vector_type(2))) float v2f;
typedef __attribute__((ext_vector_type(8))) float v8f;

#define IMG_H 512
#define IMG_W 512
#define TILE 32
#define HALO 5
#define NPLANES 48           // 16 * 3
#define TILES_PER_PLANE 256  // (512/32)^2
#define NBLOCKS (NPLANES * TILES_PER_PLANE)

__device__ __forceinline__ int reflect_idx(int p, int n) {
    // jnp.pad mode='reflect' (mirror, edge not repeated)
    if (p < 0) p = -p;
    if (p >= n) p = 2 * n - 2 - p;
    return p;
}

__global__ __launch_bounds__(128) void ssim_tile_kernel(
    const float* __restrict__ img1, const float* __restrict__ img2,
    float* __restrict__ partial) {
    // input tiles (42x42 used, padded to 48x44; rows 16B-aligned, 176B stride -> v2f loads 8B-aligned)
    __shared__ __align__(16) float sIn1[48][44];
    __shared__ __align__(16) float sIn2[48][44];
    // horizontally-blurred quantities: x, y, xx, yy, xy
    __shared__ __align__(16) float sH[5][48][32];
    __shared__ float sRed[128];

    const int tid   = threadIdx.x;
    const int lane  = tid & 31;
    const int wv    = tid >> 5;
    const int lhalf = lane >> 4;   // 0 for lanes 0-15, 1 for lanes 16-31
    const int lmod  = lane & 15;

    const int block  = blockIdx.x;
    const int plane  = block >> 8;
    const int tileid = block & 255;
    const int r0 = (tileid >> 4) * TILE;
    const int c0 = (tileid & 15) * TILE;
    const float* p1 = img1 + (size_t)plane * (IMG_H * IMG_W);
    const float* p2 = img2 + (size_t)plane * (IMG_H * IMG_W);

    // --- exact gaussian weights (11 taps, sigma=1.5), computed in registers ---
    float norm = 0.f;
#pragma unroll
    for (int i = 0; i < 11; ++i) {
        float x = (float)i - 5.0f;
        norm += expf(-(x * x) / 4.5f);
    }
    const float invnorm = 1.0f / norm;

    // Coefficient fragments of the banded matrix G[j][c] = g[j-c], j in [0,28), c = lane%16.
    // Layout (per ISA): fragment VGPR v of chunk kc holds K-index j = 4*kc + 2*(lane/16) + v.
    // Same per-lane values serve as B-operand (horizontal pass) and A-operand (vertical pass).
    v2f coef[7];
#pragma unroll
    for (int kc = 0; kc < 7; ++kc) {
#pragma unroll
        for (int v = 0; v < 2; ++v) {
            int j = 4 * kc + 2 * lhalf + v;
            int d = j - lmod;
            float val = 0.0f;
            if (d >= 0 && d <= 10) {
                float x = (float)d - 5.0f;
                val = expf(-(x * x) / 4.5f) * invnorm;
            }
            coef[kc][v] = val;
        }
    }

    // --- cooperative load: reflect-padded, clipped input tiles ---
    for (int idx = tid; idx < 48 * 44; idx += 128) {
        int row = idx / 44, col = idx % 44;
        float a = 0.0f, b = 0.0f;
        if (row < 42 && col < 42) {
            int gr = reflect_idx(r0 - HALO + row, IMG_H);
            int gc = reflect_idx(c0 - HALO + col, IMG_W);
            a = p1[gr * IMG_W + gc];
            b = p2[gr * IMG_W + gc];
            a = fminf(fmaxf(a, 0.0f), 1.0f);
            b = fminf(fmaxf(b, 0.0f), 1.0f);
        }
        sIn1[row][col] = a;
        sIn2[row][col] = b;
    }
    __syncthreads();

    // --- horizontal pass: 6 subtiles (3 rowgroups x 2 colgroups), wave-uniform loop ---
    for (int s = wv; s < 6; s += 4) {
        const int rg = s >> 1, cg = s & 1;
        v8f acc0 = {}, acc1 = {}, acc2 = {}, acc3 = {}, acc4 = {};
        const int arow = rg * 16 + lmod;
#pragma unroll
        for (int kc = 0; kc < 7; ++kc) {
            const int acol = cg * 16 + 4 * kc + 2 * lhalf;  // even -> 8B aligned
            v2f ax = *(const v2f*)&sIn1[arow][acol];
            v2f ay = *(const v2f*)&sIn2[arow][acol];
            v2f axx = ax * ax;
            v2f ayy = ay * ay;
            v2f axy = ax * ay;
            acc0 = __builtin_amdgcn_wmma_f32_16x16x4_f32(false, ax,  false, coef[kc], (short)0, acc0, false, false);
            acc1 = __builtin_amdgcn_wmma_f32_16x16x4_f32(false, ay,  false, coef[kc], (short)0, acc1, false, false);
            acc2 = __builtin_amdgcn_wmma_f32_16x16x4_f32(false, axx, false, coef[kc], (short)0, acc2, false, false);
            acc3 = __builtin_amdgcn_wmma_f32_16x16x4_f32(false, ayy, false, coef[kc], (short)0, acc3, false, false);
            acc4 = __builtin_amdgcn_wmma_f32_16x16x4_f32(false, axy, false, coef[kc], (short)0, acc4, false, false);
        }
        // scatter D (VGPR i: lanes 0-15 -> row i, lanes 16-31 -> row i+8)
        const int dcol = cg * 16 + lmod;
#pragma unroll
        for (int i = 0; i < 8; ++i) {
            const int drow = rg * 16 + i + 8 * lhalf;
            sH[0][drow][dcol] = acc0[i];
            sH[1][drow][dcol] = acc1[i];
            sH[2][drow][dcol] = acc2[i];
            sH[3][drow][dcol] = acc3[i];
            sH[4][drow][dcol] = acc4[i];
        }
    }
    __syncthreads();

    // --- vertical pass: one 16x16 subtile per wave ---
    float lsum;
    {
        const int rg = wv >> 1, cg = wv & 1;
        v8f acc[5] = {{}, {}, {}, {}, {}};
        const int bcol = cg * 16 + lmod;
#pragma unroll
        for (int kc = 0; kc < 7; ++kc) {
            const int jb = rg * 16 + 4 * kc + 2 * lhalf;
#pragma unroll
            for (int q = 0; q < 5; ++q) {
                v2f b;
                b[0] = sH[q][jb][bcol];
                b[1] = sH[q][jb + 1][bcol];
                acc[q] = __builtin_amdgcn_wmma_f32_16x16x4_f32(false, coef[kc], false, b, (short)0, acc[q], false, false);
            }
        }

        // --- SSIM map + per-lane partial loss sum ---
        const float C1 = 0.0001f;  // (0.01)^2
        const float C2 = 0.0009f;  // (0.03)^2
        lsum = 0.0f;
#pragma unroll
        for (int i = 0; i < 8; ++i) {
            float mu1 = acc[0][i], mu2 = acc[1][i];
            float e11 = acc[2][i], e22 = acc[3][i], e12 = acc[4][i];
            float mu1s = mu1 * mu1, mu2s = mu2 * mu2, mu12 = mu1 * mu2;
            float s11 = e11 - mu1s;
            float s22 = e22 - mu2s;
            float s12 = e12 - mu12;
            float num = (2.0f * mu12 + C1) * (2.0f * s12 + C2);
            float den = (mu1s + mu2s + C1) * (s11 + s22 + C2);
            float ssim = num / den;
            float loss = fminf(fmaxf((1.0f - ssim) * 0.5f, 0.0f), 1.0f);
            lsum += loss;
        }
    }
    sRed[tid] = lsum;
    __syncthreads();
    if (tid == 0) {
        float t = 0.0f;
        for (int i = 0; i < 128; ++i) t += sRed[i];
        partial[block] = t;
    }
}

__global__ __launch_bounds__(256) void ssim_reduce_kernel(
    const float* __restrict__ partial, int n, float* __restrict__ out) {
    __shared__ float s[256];
    float t = 0.0f;
    for (int i = threadIdx.x; i < n; i += 256) t += partial[i];
    s[threadIdx.x] = t;
    __syncthreads();
    for (int stride = 128; stride > 0; stride >>= 1) {
        if (threadIdx.x < stride) s[threadIdx.x] += s[threadIdx.x + stride];
        __syncthreads();
    }
    if (threadIdx.x == 0)
        out[0] = s[0] * (1.0f / (16.0f * 3.0f * 512.0f * 512.0f));
}

extern "C" void kernel_launch(void* const* d_in, const int* in_sizes, int n_in,
                              void* d_out, int out_size, void* d_ws, size_t ws_size,
                              hipStream_t stream) {
    const float* img1 = (const float*)d_in[0];  // gen_srgb01
    const float* img2 = (const float*)d_in[1];  // tgt_srgb01
    float* out = (float*)d_out;
    float* partial = (float*)d_ws;  // NBLOCKS floats of scratch

    ssim_tile_kernel<<<NBLOCKS, 128, 0, stream>>>(img1, img2, partial);
    ssim_reduce_kernel<<<1, 256, 0, stream>>>(partial, NBLOCKS, out);
}